// RepresentativeVectors_78675210928620
// MI455X (gfx1250) — compile-verified
//
#include <hip/hip_runtime.h>

typedef __attribute__((ext_vector_type(2))) float v2f;
typedef __attribute__((ext_vector_type(8))) float v8f;

#define B_    16
#define C_    64
#define N_    65536
#define NBVEC 4
#define PIX   128            // pixels per block in distance kernel
#define BPB   (N_ / PIX)     // 512 blocks per batch
#define XSTR  132            // padded LDS stride (avoid 64-bank conflicts)

// workspace layout (float indices)
#define WS_SCORE 0
#define WS_RAW   (B_ * N_)                       // B*C raw vectors
#define WS_PSUM  (WS_RAW + B_ * C_)              // B*BPB partial sim sums
#define WS_PNUM  (WS_PSUM + B_ * BPB)            // B*BPB*C partial numerators
#define WS_IND   (WS_PNUM + B_ * BPB * C_)       // B ints

// output layout (float indices): vecList | simList | selectedPos
#define OUT_VEC 0
#define OUT_SIM (B_ * NBVEC * C_)
#define OUT_POS (OUT_SIM + B_ * NBVEC * N_)

__global__ void rv_init_kernel(const float* __restrict__ score_init,
                               float* __restrict__ ws, float* __restrict__ out) {
    int i = blockIdx.x * blockDim.x + threadIdx.x;
    if (i < B_ * N_) {
        ws[WS_SCORE + i] = score_init[i];
        out[OUT_POS + i] = 0.0f;   // selectedPos is all zeros in this branch
    }
}

// one block per batch: greedy argmax over score (first-max tie break), gather raw vec
__global__ void rv_argmax_kernel(const float* __restrict__ x, float* __restrict__ ws) {
    int b = blockIdx.x, tid = threadIdx.x;     // 256 threads
    const float* sc = ws + WS_SCORE + (size_t)b * N_;
    float bv = -1.0f; int bi = 0;
    for (int n = tid; n < N_; n += 256) {
        float v = sc[n];
        if (v > bv) { bv = v; bi = n; }        // ascending scan -> first max kept
    }
    __shared__ float mv[256]; __shared__ int mi[256];
    mv[tid] = bv; mi[tid] = bi; __syncthreads();
    for (int s = 128; s > 0; s >>= 1) {
        if (tid < s) {
            float ov = mv[tid + s]; int oi = mi[tid + s];
            if (ov > mv[tid] || (ov == mv[tid] && oi < mi[tid])) { mv[tid] = ov; mi[tid] = oi; }
        }
        __syncthreads();
    }
    __shared__ int indS;
    if (tid == 0) { indS = mi[0]; ((int*)ws)[WS_IND + b] = mi[0]; }
    __syncthreads();
    if (tid < C_)
        ws[WS_RAW + b * C_ + tid] = x[((size_t)(b * C_ + tid)) * N_ + indS];
}

// distance + sim + score update + WMMA weighted-sum partials
__global__ void rv_dist_kernel(const float* __restrict__ x, float* __restrict__ ws,
                               float* __restrict__ out, int iter) {
    const int blk = blockIdx.x, b = blockIdx.y;
    const int tid = threadIdx.x;               // 128 threads = 4 waves (wave32)
    const int lane = tid & 31, wave = tid >> 5;
    const int n0 = blk * PIX;

    __shared__ float xs[C_ * XSTR];            // staged tile: xs[c][pix]
    __shared__ float rawS[C_];
    __shared__ float simbuf[PIX];
    __shared__ float numS[C_];
    __shared__ float red[PIX];

    if (tid < C_) rawS[tid] = ws[WS_RAW + b * C_ + tid];
    __syncthreads();

    // pass 1: fully coalesced stream of x; stage into LDS, accumulate d2
    const float* xb = x + (size_t)b * C_ * N_;
    float d2 = 0.0f;
    #pragma unroll 8
    for (int c = 0; c < C_; ++c) {
        float v = xb[(size_t)c * N_ + n0 + tid];
        xs[c * XSTR + tid] = v;
        float t = v - rawS[c];
        d2 += t * t;
    }
    float sim = __expf(-__fsqrt_rn(fmaxf(d2, 1e-12f)) * (1.0f / 20.0f));
    out[OUT_SIM + ((size_t)(b * NBVEC + iter)) * N_ + n0 + tid] = sim;
    ws[WS_SCORE + (size_t)b * N_ + n0 + tid] *= (1.0f - sim);
    simbuf[tid] = sim;
    red[tid] = sim;
    __syncthreads();

    // pass 2 (WMMA): num[c] += sim[n]*xf[n][c] as D(16x16) = A(16x4) x B(4x16) over 32 chunks.
    // A lane map (ISA 32-bit 16x4): lane L -> M = L%16, VGPR v -> K = 2*(L/16)+v.
    // B = sim chunk broadcast across all 16 columns, so only column 0 of D is consumed.
    const int cg = wave;                       // each wave owns one 16-channel group
    const int m  = lane & 15;
    const int kk = (lane >> 4) << 1;
    v8f acc = {};
    for (int ch = 0; ch < PIX / 4; ++ch) {
        v2f a, bb;
        a.x  = xs[(cg * 16 + m) * XSTR + ch * 4 + kk];
        a.y  = xs[(cg * 16 + m) * XSTR + ch * 4 + kk + 1];
        bb.x = simbuf[ch * 4 + kk];
        bb.y = simbuf[ch * 4 + kk + 1];
        acc = __builtin_amdgcn_wmma_f32_16x16x4_f32(
                  false, a, false, bb, (short)0, acc, false, false);
    }
    // D layout: VGPR r, lanes 0-15 -> M=r (N=lane); lanes 16-31 -> M=r+8 (N=lane-16).
    if (lane == 0) {
        #pragma unroll
        for (int r = 0; r < 8; ++r) numS[cg * 16 + r] = acc[r];
    }
    if (lane == 16) {
        #pragma unroll
        for (int r = 0; r < 8; ++r) numS[cg * 16 + 8 + r] = acc[r];
    }
    __syncthreads();

    // block sim-sum (deterministic tree)
    for (int s = PIX / 2; s > 0; s >>= 1) {
        if (tid < s) red[tid] += red[tid + s];
        __syncthreads();
    }
    const int gblk = b * BPB + blk;
    if (tid == 0) ws[WS_PSUM + gblk] = red[0];
    if (tid < C_) ws[WS_PNUM + (size_t)gblk * C_ + tid] = numS[tid];
}

// per-batch final reduction: simSum, numerator, vecList = num/simSum
__global__ void rv_final_kernel(float* __restrict__ ws, float* __restrict__ out, int iter) {
    int b = blockIdx.x, tid = threadIdx.x;     // 256 threads
    __shared__ float red[256];
    red[tid] = ws[WS_PSUM + b * BPB + tid] + ws[WS_PSUM + b * BPB + 256 + tid];
    __syncthreads();
    for (int s = 128; s > 0; s >>= 1) {
        if (tid < s) red[tid] += red[tid + s];
        __syncthreads();
    }
    __shared__ float ssum;
    if (tid == 0) ssum = red[0];
    __syncthreads();
    if (tid < C_) {
        const float* pn = ws + WS_PNUM + (size_t)b * BPB * C_;
        float acc = 0.0f;
        for (int k = 0; k < BPB; ++k) acc += pn[k * C_ + tid];
        out[OUT_VEC + (b * NBVEC + iter) * C_ + tid] = acc / ssum;
    }
}

extern "C" void kernel_launch(void* const* d_in, const int* in_sizes, int n_in,
                              void* d_out, int out_size, void* d_ws, size_t ws_size,
                              hipStream_t stream) {
    const float* x          = (const float*)d_in[0];
    const float* score_init = (const float*)d_in[1];
    float* out = (float*)d_out;
    float* ws  = (float*)d_ws;

    rv_init_kernel<<<(B_ * N_ + 255) / 256, 256, 0, stream>>>(score_init, ws, out);
    for (int t = 0; t < NBVEC; ++t) {
        rv_argmax_kernel<<<B_, 256, 0, stream>>>(x, ws);
        rv_dist_kernel<<<dim3(BPB, B_), PIX, 0, stream>>>(x, ws, out, t);
        rv_final_kernel<<<B_, 256, 0, stream>>>(ws, out, t);
    }
}